// WassersteinLog_34591666602827
// MI455X (gfx1250) — compile-verified
//
#include <hip/hip_runtime.h>
#include <math.h>

// ---------------------------------------------------------------------------
// WassersteinLog velocity on 3x3 SPD Gaussians, MI455X (gfx1250, wave32).
// VALU-bound: per-lane branchless 3x3 cyclic Jacobi (4 sweeps) using raw
// v_rcp/v_rsq/v_sqrt, symmetric-storage spectral reconstructions.
// Covariances staged via GLOBAL_LOAD_ASYNC_TO_LDS_B128 (ASYNCcnt path).
// ---------------------------------------------------------------------------

#ifndef USE_ASYNC_LDS
#define USE_ASYNC_LDS 1
#endif

#define BLK 256

#define K_MAX_COV 1.0e4f
#define K_MIN_COV 1.0e-4f
#define K_SYMM_THRESH 1.0e-5f
#define K_DET_THRESH 1.0e-6f
#define K_MAX_POS 1.0e3f
#define K_EIG_CLAMP 1.0e-4f
#define K_INV_EPS 1.0e-4f

// symmetric 3x3 storage: [00,01,02,11,12,22]
__device__ __forceinline__ constexpr int sym_idx(int i, int j) {
  return (i <= j) ? ((i == 0) ? j : ((i == 1) ? 2 + j : 5))
                  : sym_idx(j, i);
}
#define SYM(S, i, j) (S)[sym_idx((i), (j))]

// One branchless Jacobi rotation in plane (p,q). (arp,arq) = coupled pair of
// the third row; (qpX,qqX) = rows of eigenvector matrix, columns p and q.
__device__ __forceinline__ void jrot(float& app, float& aqq, float& apq,
                                     float& arp, float& arq,
                                     float& qp0, float& qq0,
                                     float& qp1, float& qq1,
                                     float& qp2, float& qq2) {
  float a = apq;
  float tau = (aqq - app) * (0.5f * __builtin_amdgcn_rcpf(a));
  float t = copysignf(
      __builtin_amdgcn_rcpf(fabsf(tau) +
                            __builtin_amdgcn_sqrtf(fmaf(tau, tau, 1.0f))),
      tau);
  // a==0 -> tau inf/NaN -> t NaN/0; discard. huge tau -> t==0 (negligible rot)
  t = (a == 0.0f) ? 0.0f : t;
  float c = __builtin_amdgcn_rsqf(fmaf(t, t, 1.0f));
  float s = t * c;
  app = fmaf(-t, a, app);
  aqq = fmaf(t, a, aqq);
  apq = 0.0f;
  float rp = arp, rq = arq;
  arp = fmaf(c, rp, -s * rq);
  arq = fmaf(s, rp, c * rq);
  float x;
  x = qp0; qp0 = fmaf(c, x, -s * qq0); qq0 = fmaf(s, x, c * qq0);
  x = qp1; qp1 = fmaf(c, x, -s * qq1); qq1 = fmaf(s, x, c * qq1);
  x = qp2; qp2 = fmaf(c, x, -s * qq2); qq2 = fmaf(s, x, c * qq2);
}

// 3x3 symmetric eigendecomposition, cyclic Jacobi, 4 sweeps (12 rotations).
// A = Q diag(lam) Q^T, Q row-major, columns are eigenvectors.
__device__ __forceinline__ void eigh3(float a00, float a01, float a02,
                                      float a11, float a12, float a22,
                                      float Q[9], float lam[3]) {
  Q[0] = 1.f; Q[1] = 0.f; Q[2] = 0.f;
  Q[3] = 0.f; Q[4] = 1.f; Q[5] = 0.f;
  Q[6] = 0.f; Q[7] = 0.f; Q[8] = 1.f;
#pragma unroll
  for (int sw = 0; sw < 4; ++sw) {
    jrot(a00, a11, a01, a02, a12, Q[0], Q[1], Q[3], Q[4], Q[6], Q[7]); // (0,1)
    jrot(a00, a22, a02, a01, a12, Q[0], Q[2], Q[3], Q[5], Q[6], Q[8]); // (0,2)
    jrot(a11, a22, a12, a01, a02, Q[1], Q[2], Q[4], Q[5], Q[7], Q[8]); // (1,2)
  }
  lam[0] = a00; lam[1] = a11; lam[2] = a22;
}

// S = Q diag(f) Q^T, symmetric storage (6 entries)
__device__ __forceinline__ void recon6(const float Q[9], float f0, float f1,
                                       float f2, float S[6]) {
  S[0] = f0 * Q[0] * Q[0] + f1 * Q[1] * Q[1] + f2 * Q[2] * Q[2];
  S[1] = f0 * Q[0] * Q[3] + f1 * Q[1] * Q[4] + f2 * Q[2] * Q[5];
  S[2] = f0 * Q[0] * Q[6] + f1 * Q[1] * Q[7] + f2 * Q[2] * Q[8];
  S[3] = f0 * Q[3] * Q[3] + f1 * Q[4] * Q[4] + f2 * Q[5] * Q[5];
  S[4] = f0 * Q[3] * Q[6] + f1 * Q[4] * Q[7] + f2 * Q[5] * Q[8];
  S[5] = f0 * Q[6] * Q[6] + f1 * Q[7] * Q[7] + f2 * Q[8] * Q[8];
}

// C = S * B  (S symmetric-6, B/C full row-major)
__device__ __forceinline__ void mm3_symL(const float S[6], const float B[9],
                                         float C[9]) {
#pragma unroll
  for (int i = 0; i < 3; ++i)
#pragma unroll
    for (int j = 0; j < 3; ++j)
      C[3 * i + j] = SYM(S, i, 0) * B[0 + j] + SYM(S, i, 1) * B[3 + j] +
                     SYM(S, i, 2) * B[6 + j];
}

// C = A * S  (A/C full row-major, S symmetric-6)
__device__ __forceinline__ void mm3_symR(const float A[9], const float S[6],
                                         float C[9]) {
#pragma unroll
  for (int i = 0; i < 3; ++i)
#pragma unroll
    for (int j = 0; j < 3; ++j)
      C[3 * i + j] = A[3 * i + 0] * SYM(S, 0, j) + A[3 * i + 1] * SYM(S, 1, j) +
                     A[3 * i + 2] * SYM(S, 2, j);
}

__global__ void __launch_bounds__(BLK)
wlog_kernel(const float* __restrict__ miu1, const float* __restrict__ miu2,
            const float* __restrict__ cov1, const float* __restrict__ cov2,
            float* __restrict__ out, int N) {
  __shared__ float s1[BLK * 9];
  __shared__ float s2[BLK * 9];

  const int tid = threadIdx.x;
  const int base = blockIdx.x * BLK;
  const int nE = min(BLK, N - base);
  const float* gc1 = cov1 + (size_t)base * 9;
  const float* gc2 = cov2 + (size_t)base * 9;

  if (nE == BLK) {
#if USE_ASYNC_LDS
    // Coalesced 16B/lane async copies, global -> LDS, bypassing VGPRs.
    // LDS byte address = low 32 bits of the generic shared pointer.
    unsigned s1b = (unsigned)(unsigned long long)(const void*)s1;
    unsigned s2b = (unsigned)(unsigned long long)(const void*)s2;
    unsigned long long g1b = (unsigned long long)(const void*)gc1;
    unsigned long long g2b = (unsigned long long)(const void*)gc2;
#pragma unroll
    for (int r = 0; r < 3; ++r) {
      int k = tid + r * BLK;                 // float4 index, 576 total
      if (k < (BLK * 9) / 4) {
        unsigned la = s1b + (unsigned)k * 16u;
        unsigned long long ga = g1b + (unsigned long long)k * 16ull;
        asm volatile("global_load_async_to_lds_b128 %0, %1, off"
                     :: "v"(la), "v"(ga) : "memory");
        unsigned lb = s2b + (unsigned)k * 16u;
        unsigned long long gb = g2b + (unsigned long long)k * 16ull;
        asm volatile("global_load_async_to_lds_b128 %0, %1, off"
                     :: "v"(lb), "v"(gb) : "memory");
      }
    }
    asm volatile("s_wait_asynccnt 0" ::: "memory");
#else
#pragma unroll
    for (int r = 0; r < 9; ++r) {
      int k = tid + r * BLK;
      s1[k] = gc1[k];
      s2[k] = gc2[k];
    }
#endif
  } else {
    for (int k = tid; k < nE * 9; k += BLK) {
      s1[k] = gc1[k];
      s2[k] = gc2[k];
    }
  }
  __syncthreads();

  const int i = base + tid;
  if (i >= N) return;

  float c1_[9], c2_[9];
#pragma unroll
  for (int k = 0; k < 9; ++k) {
    c1_[k] = s1[tid * 9 + k];
    c2_[k] = s2[tid * 9 + k];
  }
  float m1x = miu1[3 * (size_t)i + 0], m1y = miu1[3 * (size_t)i + 1],
        m1z = miu1[3 * (size_t)i + 2];
  float m2x = miu2[3 * (size_t)i + 0], m2y = miu2[3 * (size_t)i + 1],
        m2z = miu2[3 * (size_t)i + 2];

  // ---- stability mask (replicates reference semantics) ----
  float amax1 = 0.f, amin1 = 3.402823466e38f;
  float amax2 = 0.f, amin2 = 3.402823466e38f;
  float poison = 0.f;                       // stays 0 iff all entries finite
#pragma unroll
  for (int k = 0; k < 9; ++k) {
    float a1 = fabsf(c1_[k]), a2 = fabsf(c2_[k]);
    amax1 = fmaxf(amax1, a1); amin1 = fminf(amin1, a1);
    amax2 = fmaxf(amax2, a2); amin2 = fminf(amin2, a2);
    poison = fmaf(c1_[k], 0.0f, poison);    // NaN/Inf -> NaN accumulator
    poison = fmaf(c2_[k], 0.0f, poison);
  }
  bool fin = (poison == 0.0f);
  float asym1 = fmaxf(fmaxf(fabsf(c1_[1] - c1_[3]), fabsf(c1_[2] - c1_[6])),
                      fabsf(c1_[5] - c1_[7]));
  float asym2 = fmaxf(fmaxf(fabsf(c2_[1] - c2_[3]), fabsf(c2_[2] - c2_[6])),
                      fabsf(c2_[5] - c2_[7]));
  // 3x3 cofactor determinant (same expression JAX uses for 3x3 det)
  float det1 = c1_[0] * (c1_[4] * c1_[8] - c1_[5] * c1_[7]) -
               c1_[1] * (c1_[3] * c1_[8] - c1_[5] * c1_[6]) +
               c1_[2] * (c1_[3] * c1_[7] - c1_[4] * c1_[6]);
  float det2 = c2_[0] * (c2_[4] * c2_[8] - c2_[5] * c2_[7]) -
               c2_[1] * (c2_[3] * c2_[8] - c2_[5] * c2_[6]) +
               c2_[2] * (c2_[3] * c2_[7] - c2_[4] * c2_[6]);
  bool mo1 = (fabsf(m1x) < K_MAX_POS) & (fabsf(m1y) < K_MAX_POS) &
             (fabsf(m1z) < K_MAX_POS);
  bool mo2 = (fabsf(m2x) < K_MAX_POS) & (fabsf(m2y) < K_MAX_POS) &
             (fabsf(m2z) < K_MAX_POS);
  bool mask = (amax1 < K_MAX_COV) & (amax2 < K_MAX_COV) &
              (amin1 > K_MIN_COV) & (amin2 > K_MIN_COV) &
              (asym1 < K_SYMM_THRESH) & (asym2 < K_SYMM_THRESH) &
              (fabsf(det1) > K_DET_THRESH) & (fabsf(det2) > K_DET_THRESH) &
              fin & mo1 & mo2;

  // ---- substitute identity when masked (divergence-free) ----
  float s00 = mask ? c1_[0] : 1.f;
  float s11 = mask ? c1_[4] : 1.f;
  float s22 = mask ? c1_[8] : 1.f;
  float s01 = mask ? 0.5f * (c1_[1] + c1_[3]) : 0.f;
  float s02 = mask ? 0.5f * (c1_[2] + c1_[6]) : 0.f;
  float s12 = mask ? 0.5f * (c1_[5] + c1_[7]) : 0.f;
  float c2u[9];
#pragma unroll
  for (int k = 0; k < 9; ++k)
    c2u[k] = mask ? c2_[k] : ((k == 0 || k == 4 || k == 8) ? 1.f : 0.f);

  // ---- eigh(sym(cov1)); A^(1/2), regularized A^(-1/2) ----
  float Qa[9], la[3];
  eigh3(s00, s01, s02, s11, s12, s22, Qa, la);
  float f0 = sqrtf(fmaxf(la[0], K_EIG_CLAMP));
  float f1 = sqrtf(fmaxf(la[1], K_EIG_CLAMP));
  float f2 = sqrtf(fmaxf(la[2], K_EIG_CLAMP));
  float i0 = 1.0f / (K_INV_EPS + f0);
  float i1 = 1.0f / (K_INV_EPS + f1);
  float i2 = 1.0f / (K_INV_EPS + f2);
  float As[6], Ai[6];
  recon6(Qa, f0, f1, f2, As);
  recon6(Qa, i0, i1, i2, Ai);

  // ---- C = sym(A^(1/2) cov2 A^(1/2)); C^(1/2) via eigh ----
  float T[9], Cf[9];
  mm3_symL(As, c2u, T);
  mm3_symR(T, As, Cf);
  float Qc[9], lc[3];
  eigh3(Cf[0], 0.5f * (Cf[1] + Cf[3]), 0.5f * (Cf[2] + Cf[6]),
        Cf[4], 0.5f * (Cf[5] + Cf[7]), Cf[8], Qc, lc);
  float h0 = sqrtf(sqrtf(lc[0] * lc[0]));   // (l^2)^0.25 as in reference
  float h1 = sqrtf(sqrtf(lc[1] * lc[1]));
  float h2 = sqrtf(sqrtf(lc[2] * lc[2]));
  float Cs[6];
  recon6(Qc, h0, h1, h2, Cs);

  // ---- cov_vel = A^(1/2) C^(1/2) A^(-1/2); symmetrize; -2*sym(cov1) ----
  float U[9], R[9];
#pragma unroll
  for (int a = 0; a < 3; ++a)
#pragma unroll
    for (int b = 0; b < 3; ++b)
      U[3 * a + b] = SYM(As, a, 0) * SYM(Cs, 0, b) +
                     SYM(As, a, 1) * SYM(Cs, 1, b) +
                     SYM(As, a, 2) * SYM(Cs, 2, b);
  mm3_symR(U, Ai, R);

  out[3 * (size_t)i + 0] = mask ? (m2x - m1x) : 0.f;
  out[3 * (size_t)i + 1] = mask ? (m2y - m1y) : 0.f;
  out[3 * (size_t)i + 2] = mask ? (m2z - m1z) : 0.f;

  float* oc = out + (size_t)N * 3 + (size_t)i * 9;
#pragma unroll
  for (int a = 0; a < 3; ++a)
#pragma unroll
    for (int b = 0; b < 3; ++b) {
      float v = R[3 * a + b] + R[3 * b + a] - c1_[3 * a + b] - c1_[3 * b + a];
      oc[3 * a + b] = mask ? v : 0.f;
    }
}

extern "C" void kernel_launch(void* const* d_in, const int* in_sizes, int n_in,
                              void* d_out, int out_size, void* d_ws,
                              size_t ws_size, hipStream_t stream) {
  const float* miu1 = (const float*)d_in[0];
  const float* miu2 = (const float*)d_in[1];
  const float* cov1 = (const float*)d_in[2];
  const float* cov2 = (const float*)d_in[3];
  float* out = (float*)d_out;
  const int N = in_sizes[0] / 3;
  const int grid = (N + BLK - 1) / BLK;
  hipLaunchKernelGGL(wlog_kernel, dim3(grid), dim3(BLK), 0, stream,
                     miu1, miu2, cov1, cov2, out, N);
}